// GeometricAttention_22179211116601
// MI455X (gfx1250) — compile-verified
//
#include <hip/hip_runtime.h>
#include <hip/hip_bf16.h>
#include <math.h>

// Geometric attention, gfx1250 (MI455X).
// B=8, N=384, D_IN=D_MODEL=512.
// Strategy:
//  * q/k GEMMs and the scores GEMM use v_wmma_f32_16x16x4_f32 (fp32 WMMA).
//  * The O(B*N^3) masked einsum (with its ~N^3-element mask tensor input) is
//    replaced by per-row prefix sums: the mask selects two contiguous k-ranges
//    per (i,j), so sum_k M[i,j,k]*L[b,i,k] = two range sums of cumsum(L[b,i,:]).
//    d_in[12] (masking_matrix) is never read.

#define BB 8
#define NN 384
#define DD 512
#define ROWS (BB * NN)       // 3072

typedef float v2f __attribute__((ext_vector_type(2)));
typedef float v8f __attribute__((ext_vector_type(8)));

// ---------------------------------------------------------------------------
// Kernel 1: Q = h @ Wq^T + bq ; K = h @ Wk^T        (blockIdx.z: 0 -> Q, 1 -> K)
// One wave computes one 16x16 output tile, K-loop of 128 x wmma_f32_16x16x4_f32.
// A layout (16x4 f32): lane m=lane&15 holds A[m, k0], A[m, k0+1], k0=(lane>>4)*2.
// B layout (4x16 f32): lane n=lane&15 holds B[k0, n], B[k0+1, n].
// Here B[k][n] = W[n][k] (W row-major (d, c)) -> contiguous float2 along k.
// ---------------------------------------------------------------------------
__global__ __launch_bounds__(128) void ga_gemm_qk(
    const float* __restrict__ h,   // (3072, 512)
    const float* __restrict__ Wq,  // (512, 512)
    const float* __restrict__ bq,  // (512)
    const float* __restrict__ Wk,  // (512, 512)
    float* __restrict__ Q,         // (3072, 512)
    float* __restrict__ K)         // (3072, 512)
{
    const int lane = threadIdx.x & 31;
    const int wv   = threadIdx.x >> 5;
    const int colTile = blockIdx.x * 4 + wv;   // 0..31
    const int rowTile = blockIdx.y;            // 0..191
    const bool isQ = (blockIdx.z == 0);

    const float* __restrict__ W = isQ ? Wq : Wk;
    float* __restrict__ O       = isQ ? Q  : K;

    const int m   = lane & 15;
    const int hl  = lane >> 4;                 // 0 or 1
    const int row0 = rowTile * 16;
    const int col0 = colTile * 16;

    const float* Arow = h + (size_t)(row0 + m) * DD;
    const float* Brow = W + (size_t)(col0 + m) * DD;

    v8f acc = {0.f, 0.f, 0.f, 0.f, 0.f, 0.f, 0.f, 0.f};
#pragma unroll 8
    for (int kc = 0; kc < DD; kc += 4) {
        const int kk = kc + hl * 2;            // even -> 8B aligned
        v2f a = *(const v2f*)(Arow + kk);
        v2f b = *(const v2f*)(Brow + kk);
        acc = __builtin_amdgcn_wmma_f32_16x16x4_f32(
            false, a, false, b, (short)0, acc, false, false);
    }

    const float bias = isQ ? bq[col0 + m] : 0.0f;
#pragma unroll
    for (int t = 0; t < 8; ++t) {
        const int row = row0 + t + 8 * hl;     // C/D layout: VGPR t -> M = t + 8*hl
        O[(size_t)row * DD + col0 + m] = acc[t] + bias;
    }
}

// ---------------------------------------------------------------------------
// Kernel 2: left = h @ wlr^T + blr ; right = h @ wrl^T + brl  (one wave per row)
// ---------------------------------------------------------------------------
__global__ __launch_bounds__(128) void ga_leftright(
    const float* __restrict__ h,
    const float* __restrict__ wlr, const float* __restrict__ blr,
    const float* __restrict__ wrl, const float* __restrict__ brl,
    float* __restrict__ left, float* __restrict__ right)
{
    const int lane = threadIdx.x & 31;
    const int wv   = threadIdx.x >> 5;
    const int row  = blockIdx.x * 4 + wv;      // 0..3071
    const float* hr = h + (size_t)row * DD;

    float sl = 0.f, sr = 0.f;
#pragma unroll 4
    for (int c = lane; c < DD; c += 32) {
        const float hv = hr[c];
        sl += hv * wlr[c];
        sr += hv * wrl[c];
    }
#pragma unroll
    for (int off = 16; off > 0; off >>= 1) {
        sl += __shfl_xor(sl, off, 32);
        sr += __shfl_xor(sr, off, 32);
    }
    if (lane == 0) {
        left[row]  = sl + blr[0];
        right[row] = sr + brl[0];
    }
}

// ---------------------------------------------------------------------------
// Kernel 3: scores = Q_b @ K_b^T, then raw = clip(alpha*s + beta*d + gamma),
// log_p -> d_out r-region, log_1mp -> workspace. One wave per 16x16 tile.
// ---------------------------------------------------------------------------
__global__ __launch_bounds__(128) void ga_scores(
    const float* __restrict__ Q, const float* __restrict__ K,
    const float* __restrict__ left, const float* __restrict__ right,
    const float* __restrict__ alpha, const float* __restrict__ beta,
    const float* __restrict__ gamma,
    float* __restrict__ logp,     // (B, N, N) = d_out r region
    float* __restrict__ l1m)      // (B, N, N) workspace
{
    const int lane = threadIdx.x & 31;
    const int wv   = threadIdx.x >> 5;
    const int jTile = blockIdx.x * 4 + wv;     // 0..23
    const int iTile = blockIdx.y;              // 0..23
    const int b     = blockIdx.z;              // 0..7

    const float* Qb = Q + (size_t)b * NN * DD;
    const float* Kb = K + (size_t)b * NN * DD;

    const int m  = lane & 15;
    const int hl = lane >> 4;
    const float* Arow = Qb + (size_t)(iTile * 16 + m) * DD;
    const float* Brow = Kb + (size_t)(jTile * 16 + m) * DD;  // B[k][n] = K[b][n][k]

    v8f acc = {0.f, 0.f, 0.f, 0.f, 0.f, 0.f, 0.f, 0.f};
#pragma unroll 8
    for (int kc = 0; kc < DD; kc += 4) {
        const int kk = kc + hl * 2;
        v2f a = *(const v2f*)(Arow + kk);
        v2f bbv = *(const v2f*)(Brow + kk);
        acc = __builtin_amdgcn_wmma_f32_16x16x4_f32(
            false, a, false, bbv, (short)0, acc, false, false);
    }

    const float al = alpha[0], be = beta[0], ga = gamma[0];
    const int j = jTile * 16 + m;
#pragma unroll
    for (int t = 0; t < 8; ++t) {
        const int i = iTile * 16 + t + 8 * hl;
        const float dv  = (j >= i) ? left[b * NN + i] : right[b * NN + i];
        float raw = al * acc[t] + be * dv + ga;
        raw = fminf(fmaxf(raw, -16.0f), 14.0f);
        const float lp = -log1pf(expf(-raw));  // log sigmoid(raw)
        const float lm = -log1pf(expf(raw));   // log(1 - sigmoid(raw))
        const size_t idx = ((size_t)(b * NN + i)) * NN + j;
        logp[idx] = lp;
        l1m[idx]  = lm;
    }
}

// ---------------------------------------------------------------------------
// Kernel 4: in-place inclusive prefix sum of log_1mp along k, one wave per row.
// ---------------------------------------------------------------------------
__global__ __launch_bounds__(128) void ga_cumsum(float* __restrict__ L)
{
    const int lane = threadIdx.x & 31;
    const int wv   = threadIdx.x >> 5;
    const int row  = blockIdx.x * 4 + wv;      // 0..3071 (= b*N + i)
    float* p = L + (size_t)row * NN;

    float carry = 0.f;
#pragma unroll
    for (int c = 0; c < NN / 32; ++c) {
        float v = p[c * 32 + lane];
#pragma unroll
        for (int off = 1; off < 32; off <<= 1) {
            const float t = __shfl_up(v, off, 32);
            if (lane >= off) v += t;
        }
        const float tot = __shfl(v, 31, 32);
        p[c * 32 + lane] = v + carry;
        carry += tot;
    }
}

// ---------------------------------------------------------------------------
// Kernel 5: r[b,i,j] = rangeSums(C[b,i,:]) + log_p[b,i,j] - 10000*(i==j)
//           amask[b,i,j] = mask[b,j] * mask[b,i]
// Geometric band:  i<j: k in (i, j) U [max(0,2i-j+1), i)
//                  i>j: k in (j, i) U (i, min(n-1, 2i-j)]
// ---------------------------------------------------------------------------
__global__ void ga_final(
    const float* __restrict__ Csum,   // (B*N, N) inclusive cumsum
    const float* __restrict__ maskv,  // (B, N)
    float* __restrict__ r,            // (B, N, N), currently holds log_p
    float* __restrict__ amask)        // (B, N, N)
{
    const int idx = blockIdx.x * blockDim.x + threadIdx.x;
    if (idx >= BB * NN * NN) return;
    const int j  = idx % NN;
    const int bi = idx / NN;          // b*N + i
    const int i  = bi % NN;
    const int b  = bi / NN;

    const float* C = Csum + (size_t)bi * NN;
    float s = 0.f;
    if (i < j) {
        if (j - 1 >= i + 1) s += C[j - 1] - C[i];          // (i, j)
        int lo = 2 * i - j + 1; if (lo < 0) lo = 0;
        const int hi = i - 1;
        if (hi >= lo) s += C[hi] - (lo > 0 ? C[lo - 1] : 0.f);  // [2i-j+1, i)
    } else if (i > j) {
        if (i - 1 >= j + 1) s += C[i - 1] - C[j];          // (j, i)
        int hi = 2 * i - j; if (hi > NN - 1) hi = NN - 1;
        if (hi >= i + 1) s += C[hi] - C[i];                // (i, 2i-j]
    }

    float val = r[idx] + s;
    if (i == j) val -= 10000.0f;
    r[idx] = val;
    amask[idx] = maskv[b * NN + j] * maskv[b * NN + i];
}

// ---------------------------------------------------------------------------
extern "C" void kernel_launch(void* const* d_in, const int* in_sizes, int n_in,
                              void* d_out, int out_size, void* d_ws, size_t ws_size,
                              hipStream_t stream)
{
    const float* h     = (const float*)d_in[0];
    const float* mask  = (const float*)d_in[1];
    const float* Wq    = (const float*)d_in[2];
    const float* bq    = (const float*)d_in[3];
    const float* Wk    = (const float*)d_in[4];
    const float* wlr   = (const float*)d_in[5];
    const float* blr   = (const float*)d_in[6];
    const float* wrl   = (const float*)d_in[7];
    const float* brl   = (const float*)d_in[8];
    const float* alpha = (const float*)d_in[9];
    const float* beta  = (const float*)d_in[10];
    const float* gamma = (const float*)d_in[11];
    // d_in[12] = masking_matrix (N^3 bools): intentionally unused; the band
    // structure is reproduced analytically via prefix sums.

    float* ws    = (float*)d_ws;
    float* Q     = ws;                       // 3072*512
    float* K     = Q + (size_t)ROWS * DD;    // 3072*512
    float* left  = K + (size_t)ROWS * DD;    // 3072
    float* right = left + ROWS;              // 3072
    float* L     = right + ROWS;             // 8*384*384 (log_1mp -> cumsum)

    float* r  = (float*)d_out;               // (B, N, N)
    float* am = r + (size_t)BB * NN * NN;    // (B, N, N)

    ga_gemm_qk<<<dim3(8, 192, 2), 128, 0, stream>>>(h, Wq, bq, Wk, Q, K);
    ga_leftright<<<dim3(ROWS / 4), 128, 0, stream>>>(h, wlr, blr, wrl, brl, left, right);
    ga_scores<<<dim3(6, 24, 8), 128, 0, stream>>>(Q, K, left, right,
                                                  alpha, beta, gamma, r, L);
    ga_cumsum<<<dim3(ROWS / 4), 128, 0, stream>>>(L);
    const int total = BB * NN * NN;
    ga_final<<<dim3((total + 255) / 256), 256, 0, stream>>>(L, mask, r, am);
}